// RelateNet_61349312856385
// MI455X (gfx1250) — compile-verified
//
#include <hip/hip_runtime.h>
#include <math.h>

typedef __attribute__((ext_vector_type(2))) float v2f;
typedef __attribute__((ext_vector_type(8))) float v8f;

#define NB  4
#define NN  256
#define FC  512
#define GC  256
#define HD  128

// ---------------------------------------------------------------------------
// Kernel 0: uf[h] = sum_k Cf[k,h]*a_w[k]; ug likewise; c0 = (cf_b+cg_b).a_w
// ---------------------------------------------------------------------------
__global__ __launch_bounds__(HD) void prep_kernel(
    const float* __restrict__ Cf, const float* __restrict__ cfb,
    const float* __restrict__ Cg, const float* __restrict__ cgb,
    const float* __restrict__ aw,
    float* __restrict__ uf, float* __restrict__ ug, float* __restrict__ c0) {
  const int h = threadIdx.x;
  float sf = 0.f, sg = 0.f;
  for (int k = 0; k < HD; ++k) {
    const float a = aw[k];
    sf = fmaf(Cf[k * HD + h], a, sf);
    sg = fmaf(Cg[k * HD + h], a, sg);
  }
  uf[h] = sf;
  ug[h] = sg;
  if (h == 0) {
    float c = 0.f;
    for (int k = 0; k < HD; ++k) c = fmaf(cfb[k] + cgb[k], aw[k], c);
    c0[0] = c;
  }
}

// ---------------------------------------------------------------------------
// Kernel 1: projection  P = X @ W + bias  via V_WMMA_F32_16X16X4_F32.
// One wave per 16x16 output tile. Also emits A = P*u (broadcast over rows)
// and P2 = P*P for the downstream bilinear-form matmuls.
// X: [rows, K] row-major, W: [K, HD] row-major, outputs [rows, HD].
// ---------------------------------------------------------------------------
__global__ __launch_bounds__(32) void proj_kernel(
    const float* __restrict__ X, const float* __restrict__ W,
    const float* __restrict__ bias, const float* __restrict__ u,
    float* __restrict__ P, float* __restrict__ A, float* __restrict__ P2,
    int K) {
  const int lane = threadIdx.x;
  const int r    = lane & 15;     // row within A-tile / col within B-tile
  const int kh   = lane >> 4;     // K half select (lanes 16-31 -> K=2,3)
  const int m0   = blockIdx.x * 16;
  const int j0   = blockIdx.y * 16;

  v8f acc = {};
  for (int k0 = 0; k0 < K; k0 += 4) {
    const int kb = k0 + 2 * kh;
    // A tile 16x4: lane holds X[m0+r, kb] and X[m0+r, kb+1]  (aligned float2)
    v2f a = *(const v2f*)(X + (size_t)(m0 + r) * K + kb);
    // B tile 4x16: lane holds W[kb, j0+r] and W[kb+1, j0+r]
    v2f b;
    b.x = W[(size_t)kb * HD + j0 + r];
    b.y = W[(size_t)(kb + 1) * HD + j0 + r];
    acc = __builtin_amdgcn_wmma_f32_16x16x4_f32(false, a, false, b,
                                                (short)0, acc, false, false);
  }

  const int   col = j0 + r;
  const float uv  = u[col];
  const float bv  = bias[col];
#pragma unroll
  for (int t = 0; t < 8; ++t) {
    const int row = m0 + t + 8 * kh;          // C/D layout: lanes>=16 -> M+8
    const float v = acc[t] + bv;
    P [(size_t)row * HD + col] = v;
    A [(size_t)row * HD + col] = v * uv;
    P2[(size_t)row * HD + col] = v * v;
  }
}

// ---------------------------------------------------------------------------
// Kernel 2: per (b, n-tile, m-tile): four bilinear forms over H via WMMA:
//   qf = (pf*uf) @ pf^T,  rf = pf^2 @ (pf^2)^T   (same for g)
// then att = qf/max(sqrt(rf),eps) + qg/max(sqrt(rg),eps) + c0 + mask
// ---------------------------------------------------------------------------
__global__ __launch_bounds__(32) void pair_kernel(
    const float* __restrict__ pf, const float* __restrict__ af,
    const float* __restrict__ p2f,
    const float* __restrict__ pg, const float* __restrict__ ag,
    const float* __restrict__ p2g,
    const float* __restrict__ adj, const float* __restrict__ c0,
    float* __restrict__ att) {
  const int lane = threadIdx.x;
  const int r    = lane & 15;
  const int kh   = lane >> 4;
  const int m0   = blockIdx.x * 16;
  const int n0   = blockIdx.y * 16;
  const int b    = blockIdx.z;
  const size_t base = (size_t)b * NN * HD;

  const float* PF = pf  + base;
  const float* AF = af  + base;
  const float* QF = p2f + base;
  const float* PG = pg  + base;
  const float* AG = ag  + base;
  const float* QG = p2g + base;

  v8f qf = {}, rf = {}, qg = {}, rg = {};
  for (int k0 = 0; k0 < HD; k0 += 4) {
    const int kb = k0 + 2 * kh;
    v2f aF  = *(const v2f*)(AF + (size_t)(n0 + r) * HD + kb);
    v2f bF  = *(const v2f*)(PF + (size_t)(m0 + r) * HD + kb);
    v2f a2F = *(const v2f*)(QF + (size_t)(n0 + r) * HD + kb);
    v2f b2F = *(const v2f*)(QF + (size_t)(m0 + r) * HD + kb);
    v2f aG  = *(const v2f*)(AG + (size_t)(n0 + r) * HD + kb);
    v2f bG  = *(const v2f*)(PG + (size_t)(m0 + r) * HD + kb);
    v2f a2G = *(const v2f*)(QG + (size_t)(n0 + r) * HD + kb);
    v2f b2G = *(const v2f*)(QG + (size_t)(m0 + r) * HD + kb);
    qf = __builtin_amdgcn_wmma_f32_16x16x4_f32(false, aF,  false, bF,  (short)0, qf, false, false);
    rf = __builtin_amdgcn_wmma_f32_16x16x4_f32(false, a2F, false, b2F, (short)0, rf, false, false);
    qg = __builtin_amdgcn_wmma_f32_16x16x4_f32(false, aG,  false, bG,  (short)0, qg, false, false);
    rg = __builtin_amdgcn_wmma_f32_16x16x4_f32(false, a2G, false, b2G, (short)0, rg, false, false);
  }

  const float c0v = c0[0];
#pragma unroll
  for (int t = 0; t < 8; ++t) {
    const int n = n0 + t + 8 * kh;
    const int m = m0 + r;
    float v = qf[t] / fmaxf(sqrtf(rf[t]), 1e-12f)
            + qg[t] / fmaxf(sqrtf(rg[t]), 1e-12f) + c0v;
    const float adjv = adj[((size_t)b * NN + n) * NN + m];
    if (adjv == 0.f) v += -1e22f;
    att[((size_t)b * NN + n) * NN + m] = v;
  }
}

// ---------------------------------------------------------------------------
// Kernel 3: row softmax over m (256 cols), one block per row.
// ---------------------------------------------------------------------------
__global__ __launch_bounds__(NN) void softmax_kernel(
    const float* __restrict__ att, float* __restrict__ out) {
  __shared__ float red[NN];
  const int row = blockIdx.x;
  const int t   = threadIdx.x;
  const float v = att[(size_t)row * NN + t];

  red[t] = v;
  __syncthreads();
  for (int s = NN / 2; s > 0; s >>= 1) {
    if (t < s) red[t] = fmaxf(red[t], red[t + s]);
    __syncthreads();
  }
  const float mx = red[0];
  __syncthreads();

  const float e = expf(v - mx);
  red[t] = e;
  __syncthreads();
  for (int s = NN / 2; s > 0; s >>= 1) {
    if (t < s) red[t] += red[t + s];
    __syncthreads();
  }
  out[(size_t)row * NN + t] = e / red[0];
}

// ---------------------------------------------------------------------------
extern "C" void kernel_launch(void* const* d_in, const int* in_sizes, int n_in,
                              void* d_out, int out_size, void* d_ws, size_t ws_size,
                              hipStream_t stream) {
  const float* feat  = (const float*)d_in[0];
  const float* gfeat = (const float*)d_in[1];
  const float* adj   = (const float*)d_in[2];
  const float* Wf    = (const float*)d_in[3];
  const float* bf    = (const float*)d_in[4];
  const float* Wg    = (const float*)d_in[5];
  const float* bg    = (const float*)d_in[6];
  const float* Cf    = (const float*)d_in[7];
  const float* cfb   = (const float*)d_in[8];
  const float* Cg    = (const float*)d_in[9];
  const float* cgb   = (const float*)d_in[10];
  const float* aw    = (const float*)d_in[11];

  float* ws = (float*)d_ws;
  const size_t PH = (size_t)NB * NN * HD;   // 131072 floats per projection buf
  float* uf  = ws;
  float* ug  = ws + HD;
  float* c0  = ws + 2 * HD;
  float* pf  = ws + 512;
  float* pg  = pf  + PH;
  float* af  = pg  + PH;
  float* ag  = af  + PH;
  float* p2f = ag  + PH;
  float* p2g = p2f + PH;
  float* att = p2g + PH;                    // NB*NN*NN floats

  prep_kernel<<<1, HD, 0, stream>>>(Cf, cfb, Cg, cgb, aw, uf, ug, c0);

  dim3 gproj(NB * NN / 16, HD / 16);
  proj_kernel<<<gproj, 32, 0, stream>>>(feat,  Wf, bf, uf, pf, af, p2f, FC);
  proj_kernel<<<gproj, 32, 0, stream>>>(gfeat, Wg, bg, ug, pg, ag, p2g, GC);

  dim3 gpair(NN / 16, NN / 16, NB);
  pair_kernel<<<gpair, 32, 0, stream>>>(pf, af, p2f, pg, ag, p2g, adj, c0, att);

  softmax_kernel<<<NB * NN, NN, 0, stream>>>(att, (float*)d_out);
}